// BilinearInterpolation_1829656068636
// MI455X (gfx1250) — compile-verified
//
#include <hip/hip_runtime.h>

// ---------------------------------------------------------------------------
// Projective-warp bilinear resample (spatial-transformer sampling), MI455X.
//
// Shapes: X (16,384,384,64) f32, T (16,9) f32  ->  out (16,224,224,64) f32.
//
// Roofline: ~0.36 GFLOP vs ~0.81 GB HBM traffic -> 0.45 FLOP/B: pure
// bandwidth problem (~35 us floor at 23.3 TB/s). WMMA/TDM inapplicable
// (data-dependent gather, no shared-operand matmul). Optimized for the
// CDNA5 memory path + instruction-issue efficiency:
//   * 8 lanes x 2*float4 per pixel (4 pixels / wave32): per-pixel scalar math
//     (transform, divide, clip, weights) amortized over 1 KB of gathers;
//     every global_load_b128 covers a full 128B cache line per pixel-group.
//   * 32-bit element indices (X spans 604MB < 2^31 bytes) -> saddr + u32
//     voffset addressing, no per-lane 64-bit address arithmetic.
//   * nontemporal 128-bit stores keep the write-once 205MB output from
//     evicting the reused input band out of the 192MB L2.
//   * batch index block-uniform (50176 % 32 == 0) -> transform coefficients
//     load via s_load into SGPRs.
//   * single IEEE divide (reciprocal) instead of two.
// ---------------------------------------------------------------------------

typedef float v4f __attribute__((ext_vector_type(4)));

#define OUT_H 224
#define OUT_W 224
#define IN_H  384
#define IN_W  384
#define CH    64
#define BATCH 16

__global__ __launch_bounds__(256) void BilinearWarp_kernel(
    const float* __restrict__ X,     // (B, IN_H, IN_W, CH)
    const float* __restrict__ T,     // (B, 9)
    float* __restrict__ out)         // (B, OUT_H, OUT_W, CH)
{
    const int tid = blockIdx.x * 256 + threadIdx.x;
    const int p   = tid >> 3;            // output pixel id (8 lanes / pixel)
    const int c0  = (tid & 7) << 2;      // chunk 0: channels [c0, c0+4)
                                         // chunk 1: channels [c0+32, c0+36)

    // 224*224 = 50176 is a multiple of 32 pixels/block -> b is block-uniform,
    // so the transform loads below stay scalar (s_load).
    const int pixPerImg = OUT_H * OUT_W;
    const int b   = (blockIdx.x * 32) / pixPerImg;
    const int rem = p - b * pixPerImg;
    const int oy  = rem / OUT_W;
    const int ox  = rem - oy * OUT_W;

    const float* __restrict__ t = T + b * 9;
    const float t0 = t[0], t1 = t[1], t2 = t[2];
    const float t3 = t[3], t4 = t[4], t5 = t[5];
    const float t6 = t[6], t7 = t[7], t8 = t[8];

    // linspace(-1, 1, N): -1 + i * 2/(N-1)
    const float gx = -1.0f + (float)ox * (2.0f / (float)(OUT_W - 1));
    const float gy = -1.0f + (float)oy * (2.0f / (float)(OUT_H - 1));

    // sg = T @ [gx, gy, 1];  perspective divide via single IEEE reciprocal
    const float sx = t0 * gx + t1 * gy + t2;
    const float sy = t3 * gx + t4 * gy + t5;
    const float sz = t6 * gx + t7 * gy + t8 + 1e-6f;
    const float rz = 1.0f / sz;

    float x = sx * rz;
    float y = sy * rz;
    x = 0.5f * (x + 1.0f) * (float)IN_W;   // note: * W, not (W-1), per reference
    y = 0.5f * (y + 1.0f) * (float)IN_H;

    // trunc-toward-zero (matches tf.cast / jnp astype int32), +1 BEFORE clip
    int x0 = (int)x; int x1 = x0 + 1;
    int y0 = (int)y; int y1 = y0 + 1;
    x0 = min(max(x0, 0), IN_W - 1);
    x1 = min(max(x1, 0), IN_W - 1);
    y0 = min(max(y0, 0), IN_H - 1);
    y1 = min(max(y1, 0), IN_H - 1);

    // weights from the CLIPPED integer coords (matches reference)
    const float x0f = (float)x0, x1f = (float)x1;
    const float y0f = (float)y0, y1f = (float)y1;
    const float wa = (x1f - x) * (y1f - y);
    const float wb = (x1f - x) * (y - y0f);
    const float wc = (x - x0f) * (y1f - y);
    const float wd = (x - x0f) * (y - y0f);

    // 32-bit element offsets (max ~151M elements, byte offset < 2^31).
    const int imgBase = b * (IN_H * IN_W * CH);
    const int ra = imgBase + (y0 * IN_W + x0) * CH + c0;
    const int rb = imgBase + (y1 * IN_W + x0) * CH + c0;
    const int rc = imgBase + (y0 * IN_W + x1) * CH + c0;
    const int rd = imgBase + (y1 * IN_W + x1) * CH + c0;

    // Two 128B halves of each 256B channel row; each load instruction's
    // 8-lane pixel-group covers a full contiguous cache line.
    const v4f pa0 = *(const v4f*)(X + ra);
    const v4f pb0 = *(const v4f*)(X + rb);
    const v4f pc0 = *(const v4f*)(X + rc);
    const v4f pd0 = *(const v4f*)(X + rd);
    const v4f pa1 = *(const v4f*)(X + ra + 32);
    const v4f pb1 = *(const v4f*)(X + rb + 32);
    const v4f pc1 = *(const v4f*)(X + rc + 32);
    const v4f pd1 = *(const v4f*)(X + rd + 32);

    const v4f o0 = wa * pa0 + wb * pb0 + wc * pc0 + wd * pd0;
    const v4f o1 = wa * pa1 + wb * pb1 + wc * pc1 + wd * pd1;

    // Write-once output: non-temporal 128-bit stores.
    const int ob = (p << 6) + c0;      // p * CH + c0
    __builtin_nontemporal_store(o0, (v4f*)(out + ob));
    __builtin_nontemporal_store(o1, (v4f*)(out + ob + 32));
}

extern "C" void kernel_launch(void* const* d_in, const int* in_sizes, int n_in,
                              void* d_out, int out_size, void* d_ws, size_t ws_size,
                              hipStream_t stream) {
    const float* X = (const float*)d_in[0];   // (16,384,384,64) f32
    const float* T = (const float*)d_in[1];   // (16,9) f32
    float* out     = (float*)d_out;           // (16,224,224,64) f32

    const int totalPix = BATCH * OUT_H * OUT_W;   // 802816
    const int blocks   = totalPix / 32;           // 32 pixels per 256-thread block
    BilinearWarp_kernel<<<blocks, 256, 0, stream>>>(X, T, out);
}